// GNN_55783035240818
// MI455X (gfx1250) — compile-verified
//
#include <hip/hip_runtime.h>

typedef __attribute__((ext_vector_type(2))) float v2f;
typedef __attribute__((ext_vector_type(8))) float v8f;

#define HID 64
#define OUTC 32

// ---------------------------------------------------------------------------
// Zero-fill (graph-capture-safe replacement for hipMemsetAsync)
// ---------------------------------------------------------------------------
__global__ __launch_bounds__(256) void fill_zero_kernel(float4* __restrict__ p, long long n4) {
  long long i = blockIdx.x * 256LL + threadIdx.x;
  if (i < n4) p[i] = make_float4(0.f, 0.f, 0.f, 0.f);
}

// ---------------------------------------------------------------------------
// Degree: one f32 atomic per edge (L2-resident, 400 KB table)
// ---------------------------------------------------------------------------
__global__ __launch_bounds__(256) void degree_kernel(const int* __restrict__ dst,
                                                     float* __restrict__ deg, int n_edges) {
  int e = blockIdx.x * 256 + threadIdx.x;
  if (e < n_edges) atomicAdd(&deg[dst[e]], 1.0f);
}

// ---------------------------------------------------------------------------
// Scatter-add: 16 threads per edge, 4 channels each.
// Gather x[src] (float4, coalesced across the 16 threads of an edge) and
// atomically accumulate into agg[dst]. Both tables fit in the 192 MB L2.
// ---------------------------------------------------------------------------
__global__ __launch_bounds__(256) void scatter_add_kernel(const float* __restrict__ x,
                                                          const int* __restrict__ src,
                                                          const int* __restrict__ dst,
                                                          float* __restrict__ agg, int n_edges) {
  long long gid = blockIdx.x * 256LL + threadIdx.x;
  if (gid >= (long long)n_edges * 16) return;
  int e = (int)(gid >> 4);
  int c = ((int)gid & 15) * 4;
  int s = src[e];
  int d = dst[e];
  const float4 v = *(const float4*)(x + (long long)s * HID + c);
  float* o = agg + (long long)d * HID + c;
  atomicAdd(o + 0, v.x);
  atomicAdd(o + 1, v.y);
  atomicAdd(o + 2, v.z);
  atomicAdd(o + 3, v.w);
}

// ---------------------------------------------------------------------------
// Stage a (64 x NCOLS) row-major weight matrix into LDS in WMMA B-fragment
// order, so each (kt, nt) B fragment is a single lane-indexed ds_load_b64:
//   frag[((kt*NT + nt)*32 + lane)] = { W[k0][c], W[k0+1][c] }
//   with k0 = kt*4 + (lane>>4)*2, c = nt*16 + (lane&15)
// Inverse map used for staging element W[k][n]:
//   kt = k>>2, khalf = (k&3)>>1, kpair = k&1, nt = n>>4,
//   lane = (khalf<<4) | (n&15)
// ---------------------------------------------------------------------------
template <int NCOLS>
__device__ __forceinline__ void stage_weights_frag(const float* __restrict__ W,
                                                   float* __restrict__ sfrag) {
  constexpr int NT = NCOLS / 16;
  for (int i = threadIdx.x; i < HID * NCOLS; i += 256) {
    const int k = i / NCOLS;
    const int n = i % NCOLS;
    const int kt    = k >> 2;
    const int khalf = (k & 3) >> 1;
    const int kpair = k & 1;
    const int nt    = n >> 4;
    const int lane  = (khalf << 4) | (n & 15);
    sfrag[(((kt * NT + nt) * 32) + lane) * 2 + kpair] = W[i];
  }
}

// ---------------------------------------------------------------------------
// Fused SAGE layer: out = relu((agg/deg) @ Wl + x @ Wr + b)
// One wave32 per 16-node row tile; weights staged in LDS in fragment order.
// V_WMMA_F32_16X16X4_F32:
//   A 16x4 f32:  lanes 0-15 -> M=lane, K=0,1 ; lanes 16-31 -> M=lane-16, K=2,3
//   B  4x16 f32: lanes 0-15 -> N=lane, K=0,1 ; lanes 16-31 -> N=lane-16, K=2,3
//   C/D 16x16:   VGPR r: lanes 0-15 -> M=r, N=lane ; lanes 16-31 -> M=r+8
// ---------------------------------------------------------------------------
template <bool RELU>
__global__ __launch_bounds__(256) void sage_gemm_kernel(const float* __restrict__ xin,
                                                        const float* __restrict__ agg,
                                                        const float* __restrict__ deg,
                                                        const float* __restrict__ Wl,
                                                        const float* __restrict__ Wr,
                                                        const float* __restrict__ bias,
                                                        float* __restrict__ out, int n_nodes) {
  __shared__ float sWl[HID * HID];  // fragment-major
  __shared__ float sWr[HID * HID];  // fragment-major
  stage_weights_frag<HID>(Wl, sWl);
  stage_weights_frag<HID>(Wr, sWr);
  __syncthreads();

  const int wave = threadIdx.x >> 5;
  const int lane = threadIdx.x & 31;
  const int tile = blockIdx.x * 8 + wave;
  const int row_base = tile * 16;
  if (row_base >= n_nodes) return;  // wave-uniform: EXEC stays all-1s for WMMA

  const int mrow  = lane & 15;  // M index of this lane's A row
  const int khalf = lane >> 4;  // 0 -> K=0,1 ; 1 -> K=2,3
  const int rA    = row_base + mrow;

  const float invd = 1.0f / fmaxf(deg[rA], 1.0f);
  const float* __restrict__ ap = agg + (long long)rA * HID + khalf * 2;
  const float* __restrict__ xp = xin + (long long)rA * HID + khalf * 2;

  const v2f* __restrict__ fWl = (const v2f*)sWl;  // one b64 per B fragment
  const v2f* __restrict__ fWr = (const v2f*)sWr;

  v8f acc[4] = {};  // 4 column tiles of 16 -> 64 output channels

  for (int kt = 0; kt < 16; ++kt) {
    v2f aA, aX;
    aA[0] = ap[kt * 4] * invd;
    aA[1] = ap[kt * 4 + 1] * invd;
    aX[0] = xp[kt * 4];
    aX[1] = xp[kt * 4 + 1];
#pragma unroll
    for (int nt = 0; nt < 4; ++nt) {
      const int fidx = (kt * 4 + nt) * 32 + lane;
      const v2f bL = fWl[fidx];
      const v2f bR = fWr[fidx];
      acc[nt] = __builtin_amdgcn_wmma_f32_16x16x4_f32(false, aA, false, bL, (short)0,
                                                      acc[nt], false, false);
      acc[nt] = __builtin_amdgcn_wmma_f32_16x16x4_f32(false, aX, false, bR, (short)0,
                                                      acc[nt], false, false);
    }
  }

  const int m0 = row_base + khalf * 8;  // C layout: lanes>=16 hold M = r+8
#pragma unroll
  for (int nt = 0; nt < 4; ++nt) {
    const int c = nt * 16 + mrow;
    const float bv = bias[c];
#pragma unroll
    for (int r = 0; r < 8; ++r) {
      float v = acc[nt][r] + bv;
      if (RELU) v = fmaxf(v, 0.0f);
      out[(long long)(m0 + r) * HID + c] = v;
    }
  }
}

// ---------------------------------------------------------------------------
// Final linear: out = h @ Wlin + blin   (64 -> 32, no ReLU)
// ---------------------------------------------------------------------------
__global__ __launch_bounds__(256) void final_gemm_kernel(const float* __restrict__ h,
                                                         const float* __restrict__ W,
                                                         const float* __restrict__ bias,
                                                         float* __restrict__ out, int n_nodes) {
  __shared__ float sW[HID * OUTC];  // fragment-major
  stage_weights_frag<OUTC>(W, sW);
  __syncthreads();

  const int wave = threadIdx.x >> 5;
  const int lane = threadIdx.x & 31;
  const int tile = blockIdx.x * 8 + wave;
  const int row_base = tile * 16;
  if (row_base >= n_nodes) return;

  const int mrow  = lane & 15;
  const int khalf = lane >> 4;
  const int rA    = row_base + mrow;
  const float* __restrict__ hp = h + (long long)rA * HID + khalf * 2;
  const v2f* __restrict__ fW = (const v2f*)sW;

  v8f acc[2] = {};

  for (int kt = 0; kt < 16; ++kt) {
    v2f aX;
    aX[0] = hp[kt * 4];
    aX[1] = hp[kt * 4 + 1];
#pragma unroll
    for (int nt = 0; nt < 2; ++nt) {
      const v2f bW = fW[(kt * 2 + nt) * 32 + lane];
      acc[nt] = __builtin_amdgcn_wmma_f32_16x16x4_f32(false, aX, false, bW, (short)0,
                                                      acc[nt], false, false);
    }
  }

  const int m0 = row_base + khalf * 8;
#pragma unroll
  for (int nt = 0; nt < 2; ++nt) {
    const int c = nt * 16 + mrow;
    const float bv = bias[c];
#pragma unroll
    for (int r = 0; r < 8; ++r)
      out[(long long)(m0 + r) * OUTC + c] = acc[nt][r] + bv;
  }
}

// ---------------------------------------------------------------------------
extern "C" void kernel_launch(void* const* d_in, const int* in_sizes, int n_in,
                              void* d_out, int out_size, void* d_ws, size_t ws_size,
                              hipStream_t stream) {
  const float* x    = (const float*)d_in[0];
  const int*   ei   = (const int*)d_in[1];
  const float* Wl1  = (const float*)d_in[2];
  const float* Wr1  = (const float*)d_in[3];
  const float* b1   = (const float*)d_in[4];
  const float* Wl2  = (const float*)d_in[5];
  const float* Wr2  = (const float*)d_in[6];
  const float* b2   = (const float*)d_in[7];
  const float* Wl3  = (const float*)d_in[8];
  const float* Wr3  = (const float*)d_in[9];
  const float* b3   = (const float*)d_in[10];
  const float* Wlin = (const float*)d_in[11];
  const float* blin = (const float*)d_in[12];

  const int n_nodes = in_sizes[0] / HID;
  const int n_edges = in_sizes[1] / 2;
  const int* src = ei;
  const int* dst = ei + n_edges;

  // Workspace carve-up (all fp32): deg | agg | h1 | h2
  float* deg = (float*)d_ws;
  size_t degN = ((size_t)n_nodes + 255) & ~(size_t)255;
  float* agg = deg + degN;
  float* h1  = agg + (size_t)n_nodes * HID;
  float* h2  = h1 + (size_t)n_nodes * HID;

  const int tiles       = (n_nodes + 15) / 16;
  const int gemm_blocks = (tiles + 7) / 8;
  const long long scat  = (long long)n_edges * 16;
  const int scat_blocks = (int)((scat + 255) / 256);
  const int deg_blocks  = (n_edges + 255) / 256;

  const long long featN = (long long)n_nodes * HID;

  // degree (computed once, reused by every layer)
  fill_zero_kernel<<<(int)((degN / 4 + 255) / 256), 256, 0, stream>>>((float4*)deg, (long long)(degN / 4));
  degree_kernel<<<deg_blocks, 256, 0, stream>>>(dst, deg, n_edges);

  // layer 1
  fill_zero_kernel<<<(int)((featN / 4 + 255) / 256), 256, 0, stream>>>((float4*)agg, featN / 4);
  scatter_add_kernel<<<scat_blocks, 256, 0, stream>>>(x, src, dst, agg, n_edges);
  sage_gemm_kernel<true><<<gemm_blocks, 256, 0, stream>>>(x, agg, deg, Wl1, Wr1, b1, h1, n_nodes);

  // layer 2
  fill_zero_kernel<<<(int)((featN / 4 + 255) / 256), 256, 0, stream>>>((float4*)agg, featN / 4);
  scatter_add_kernel<<<scat_blocks, 256, 0, stream>>>(h1, src, dst, agg, n_edges);
  sage_gemm_kernel<true><<<gemm_blocks, 256, 0, stream>>>(h1, agg, deg, Wl2, Wr2, b2, h2, n_nodes);

  // layer 3
  fill_zero_kernel<<<(int)((featN / 4 + 255) / 256), 256, 0, stream>>>((float4*)agg, featN / 4);
  scatter_add_kernel<<<scat_blocks, 256, 0, stream>>>(h2, src, dst, agg, n_edges);
  sage_gemm_kernel<true><<<gemm_blocks, 256, 0, stream>>>(h2, agg, deg, Wl3, Wr3, b3, h1, n_nodes);

  // final linear -> d_out
  final_gemm_kernel<<<gemm_blocks, 256, 0, stream>>>(h1, Wlin, blin, (float*)d_out, n_nodes);
}